// RNNDecoder_5858335392394
// MI455X (gfx1250) — compile-verified
//
#include <hip/hip_runtime.h>
#include <hip/hip_bf16.h>

// ---------------------------------------------------------------------------
// LSTM greedy decoder for MI455X (gfx1250, wave32, WMMA + async-to-LDS).
// VOCAB=50257, EMBED=HIDDEN=1024, TGT_LEN=256.  All data f32.
// HBM-bound: ~61 GB of weight traffic total -> ~2.6 ms floor @ 23.3 TB/s.
// ---------------------------------------------------------------------------

#define VOCAB   50257
#define HID     1024
#define TGT     256
#define SOS_TOK 1

#define ROWS_PB 128          // vocab rows per block in logits kernel
#define NBLK_V  ((VOCAB + ROWS_PB - 1) / ROWS_PB)   // 393
#define KCH     32           // K chunk staged in LDS
#define LSTR    36           // LDS row stride (floats): conflict-free float2 reads

typedef __attribute__((ext_vector_type(2))) float v2f;
typedef __attribute__((ext_vector_type(8))) float v8f;

// ---------------------------------------------------------------------------
__global__ void lstm_init_kernel(const float* __restrict__ init_h,
                                 const float* __restrict__ init_c,
                                 float* __restrict__ h0,
                                 float* __restrict__ cbuf,
                                 int* __restrict__ tok) {
  int i = blockIdx.x * blockDim.x + threadIdx.x;
  if (i < HID) {
    h0[i]   = init_h[i];
    cbuf[i] = init_c[i];
  }
  if (i == 0) tok[0] = SOS_TOK;
}

// ---------------------------------------------------------------------------
// One block per hidden unit j.  4 waves = gate rows j, j+H, j+2H, j+3H.
// Lane-strided coalesced dot over K=1024 for W_ih@relu(E[tok]) + W_hh@h.
// Wave 0 fuses the LSTM cell update.
__global__ __launch_bounds__(128) void lstm_gates_kernel(
    const float* __restrict__ Wih, const float* __restrict__ Whh,
    const float* __restrict__ bih, const float* __restrict__ bhh,
    const float* __restrict__ emb, const int* __restrict__ tok,
    const float* __restrict__ h_in, float* __restrict__ h_out,
    float* __restrict__ cbuf) {
  __shared__ float g[4];
  const int j    = blockIdx.x;
  const int wid  = threadIdx.x >> 5;   // gate index 0..3 (i,f,g,o)
  const int lane = threadIdx.x & 31;

  const int tk = tok[0];
  const float* __restrict__ erow = emb + (size_t)tk * HID;
  const int r = wid * HID + j;
  const float* __restrict__ wi = Wih + (size_t)r * HID;
  const float* __restrict__ wh = Whh + (size_t)r * HID;

  float s = 0.f;
  #pragma unroll 4
  for (int k = lane; k < HID; k += 32) {
    float x = erow[k];
    x = x > 0.f ? x : 0.f;             // relu(embedding)
    s = fmaf(wi[k], x, s);
    s = fmaf(wh[k], h_in[k], s);
  }
  #pragma unroll
  for (int m = 16; m > 0; m >>= 1) s += __shfl_xor(s, m, 32);
  if (lane == 0) g[wid] = s + bih[r] + bhh[r];
  __syncthreads();

  if (threadIdx.x == 0) {
    const float gi = g[0], gf = g[1], gg = g[2], go = g[3];
    const float i_ = 1.f / (1.f + expf(-gi));
    const float f_ = 1.f / (1.f + expf(-gf));
    const float g_ = tanhf(gg);
    const float o_ = 1.f / (1.f + expf(-go));
    const float cn = f_ * cbuf[j] + i_ * g_;
    cbuf[j]  = cn;
    h_out[j] = o_ * tanhf(cn);
  }
}

// ---------------------------------------------------------------------------
// WMMA-based logits GEMV with async global->LDS double-buffered staging.
// Block handles 128 vocab rows, 8 waves x 16 rows.
// A (16x4) = h chunk broadcast over M; B (4x16) = W_out^T tile from LDS.
// D[m][n] == y[row base + n] for every m; lanes 0..15 read acc[0].
__global__ __launch_bounds__(256) void logits_wmma_kernel(
    const float* __restrict__ Wout, const float* __restrict__ bout,
    const float* __restrict__ h, float* __restrict__ out_row,
    float* __restrict__ pmax, float* __restrict__ psum,
    int* __restrict__ pidx) {
  __shared__ float lw[2][ROWS_PB * LSTR];  // 2 x 128 x 36 floats (36864 B)
  __shared__ float lh[2][KCH];
  __shared__ float vals[ROWS_PB];
  __shared__ float red[ROWS_PB];
  __shared__ int   redi[ROWS_PB];

  const int t    = threadIdx.x;
  const int wid  = t >> 5;       // wave 0..7
  const int lane = t & 31;
  const int r0   = blockIdx.x * ROWS_PB;

  // Stage one K-chunk: 128 rows x 32 floats of W_out via async-to-LDS b128
  // (4 instructions/wave, ASYNCcnt-tracked, no VGPR round trip), plus the
  // 32-float h chunk via a normal LDS store.
  auto stage = [&](int k0, int buf) {
    if (t < KCH) lh[buf][t] = h[k0 + t];
    #pragma unroll
    for (int p = 0; p < 4; ++p) {
      const int idx = p * 256 + t;       // 0..1023
      const int row = idx >> 3;
      const int c4  = idx & 7;
      int gr = r0 + row;
      if (gr >= VOCAB) gr = VOCAB - 1;   // clamp; masked later
      // 8 consecutive lanes cover one full 128B line -> perfect coalescing.
      const unsigned long long gp =
          (unsigned long long)(const void*)(Wout + (size_t)gr * HID + k0 + c4 * 4);
      // Flat LDS addresses carry the LDS byte offset in addr[31:0].
      const unsigned la =
          (unsigned)(unsigned long long)(const void*)&lw[buf][row * LSTR + c4 * 4];
      asm volatile("global_load_async_to_lds_b128 %0, %1, off"
                   :: "v"(la), "v"(gp)
                   : "memory");
    }
  };

  v8f acc = {};

  stage(0, 0);
  for (int k0 = 0; k0 < HID; k0 += KCH) {
    const int  buf  = (k0 / KCH) & 1;
    const bool more = (k0 + KCH) < HID;
    if (more) {
      stage(k0 + KCH, buf ^ 1);          // run copy engine one chunk ahead
      asm volatile("s_wait_asynccnt 4" ::: "memory");  // chunk k0 landed
    } else {
      asm volatile("s_wait_asynccnt 0" ::: "memory");
    }
    __syncthreads();

    const int n    = lane & 15;
    const int koff = (lane < 16) ? 0 : 2;      // K halves per ISA A/B layout
    const float* __restrict__ bp0 = &lw[buf][(wid * 16 + n) * LSTR + koff];
    const float* __restrict__ ap0 = &lh[buf][koff];
    #pragma unroll
    for (int kk = 0; kk < KCH; kk += 4) {
      v2f a, b;
      a.x = ap0[kk];
      a.y = ap0[kk + 1];
      b.x = bp0[kk];
      b.y = bp0[kk + 1];
      acc = __builtin_amdgcn_wmma_f32_16x16x4_f32(
          false, a, false, b, (short)0, acc, false, false);
    }
    __syncthreads();
  }

  // Extract GEMV result: lane l (<16) of wave w holds y[r0 + w*16 + l] in acc[0].
  const int gr = r0 + wid * 16 + (lane & 15);
  if (lane < 16) {
    float yo = -__builtin_inff();
    if (gr < VOCAB) {
      yo = acc[0] + bout[gr];
      out_row[gr] = yo;                 // raw logits; logZ subtracted later
    }
    vals[wid * 16 + (lane & 15)] = yo;
  }
  __syncthreads();

  // Deterministic block (max, argmax) tree over 128 values.
  if (t < ROWS_PB) { red[t] = vals[t]; redi[t] = r0 + t; }
  __syncthreads();
  for (int s = ROWS_PB / 2; s > 0; s >>= 1) {
    if (t < s) {
      const float a1 = red[t], a2 = red[t + s];
      const int   i1 = redi[t], i2 = redi[t + s];
      if (a2 > a1 || (a2 == a1 && i2 < i1)) { red[t] = a2; redi[t] = i2; }
    }
    __syncthreads();
  }
  const float M  = red[0];
  const int   am = redi[0];
  __syncthreads();

  // Deterministic block sum of exp(v - M).
  if (t < ROWS_PB) red[t] = expf(vals[t] - M);   // -inf -> 0
  __syncthreads();
  for (int s = ROWS_PB / 2; s > 0; s >>= 1) {
    if (t < s) red[t] += red[t + s];
    __syncthreads();
  }
  if (t == 0) {
    pmax[blockIdx.x] = M;
    psum[blockIdx.x] = red[0];
    pidx[blockIdx.x] = am;
  }
}

// ---------------------------------------------------------------------------
// Combine per-block partials: global max/argmax (first occurrence), logZ,
// next token.  Fixed-order trees -> bit-deterministic across graph replays.
__global__ __launch_bounds__(512) void softmax_reduce_kernel(
    const float* __restrict__ pmax, const float* __restrict__ psum,
    const int* __restrict__ pidx, float* __restrict__ logZ,
    int* __restrict__ tok) {
  __shared__ float vmax[512];
  __shared__ int   vidx[512];
  __shared__ float vsum[512];
  const int t = threadIdx.x;

  float m  = -__builtin_inff();
  int   id = 0x7fffffff;
  if (t < NBLK_V) { m = pmax[t]; id = pidx[t]; }
  vmax[t] = m; vidx[t] = id;
  __syncthreads();
  for (int s = 256; s > 0; s >>= 1) {
    if (t < s) {
      const float a1 = vmax[t], a2 = vmax[t + s];
      const int   i1 = vidx[t], i2 = vidx[t + s];
      if (a2 > a1 || (a2 == a1 && i2 < i1)) { vmax[t] = a2; vidx[t] = i2; }
    }
    __syncthreads();
  }
  const float M  = vmax[0];
  const int   am = vidx[0];
  __syncthreads();

  float e = 0.f;
  if (t < NBLK_V) e = psum[t] * expf(pmax[t] - M);
  vsum[t] = e;
  __syncthreads();
  for (int s = 256; s > 0; s >>= 1) {
    if (t < s) vsum[t] += vsum[t + s];
    __syncthreads();
  }
  if (t == 0) {
    logZ[0] = M + logf(vsum[0]);
    tok[0]  = am;                       // argmax(logp) == argmax(logits)
  }
}

// ---------------------------------------------------------------------------
__global__ void logp_writeout_kernel(float* __restrict__ out_row,
                                     const float* __restrict__ logZ) {
  const int i = blockIdx.x * blockDim.x + threadIdx.x;
  if (i < VOCAB) out_row[i] -= logZ[0];
}

// ---------------------------------------------------------------------------
extern "C" void kernel_launch(void* const* d_in, const int* in_sizes, int n_in,
                              void* d_out, int out_size, void* d_ws,
                              size_t ws_size, hipStream_t stream) {
  (void)in_sizes; (void)n_in; (void)out_size; (void)ws_size;

  const float* init_h = (const float*)d_in[0];
  const float* init_c = (const float*)d_in[1];
  const float* emb    = (const float*)d_in[2];
  const float* Wih    = (const float*)d_in[3];
  const float* Whh    = (const float*)d_in[4];
  const float* bih    = (const float*)d_in[5];
  const float* bhh    = (const float*)d_in[6];
  const float* Wout   = (const float*)d_in[7];
  const float* bout   = (const float*)d_in[8];
  float* out = (float*)d_out;

  // Workspace layout (floats): h ping-pong, c, partials, scalars. ~19 KB.
  float* ws_f = (float*)d_ws;
  float* h0   = ws_f;                 // 1024
  float* h1   = ws_f + 1024;          // 1024
  float* cbuf = ws_f + 2048;          // 1024
  float* pmax = ws_f + 3072;          // 512 (NBLK_V=393 used)
  float* psum = ws_f + 3584;          // 512
  float* logZ = ws_f + 4096;          // 1
  int*   pidx = (int*)(ws_f + 4128);  // 512
  int*   tok  = (int*)(ws_f + 4640);  // 1

  lstm_init_kernel<<<4, 256, 0, stream>>>(init_h, init_c, h0, cbuf, tok);

  for (int t = 0; t < TGT; ++t) {
    const float* hin = (t & 1) ? h1 : h0;
    float*       hout = (t & 1) ? h0 : h1;
    float* row = out + (size_t)t * VOCAB;

    lstm_gates_kernel<<<HID, 128, 0, stream>>>(Wih, Whh, bih, bhh, emb, tok,
                                               hin, hout, cbuf);
    logits_wmma_kernel<<<NBLK_V, 256, 0, stream>>>(Wout, bout, hout, row,
                                                   pmax, psum, pidx);
    softmax_reduce_kernel<<<1, 512, 0, stream>>>(pmax, psum, pidx, logZ, tok);
    logp_writeout_kernel<<<(VOCAB + 255) / 256, 256, 0, stream>>>(row, logZ);
  }
}